// RegularizedCFMNet_47270410060276
// MI455X (gfx1250) — compile-verified
//
#include <hip/hip_runtime.h>

// ---------------------------------------------------------------------------
// RegularizedCFMNet for MI455X (gfx1250, wave32, WMMA).
// Pipeline:
//   1) spec_project : A = sgx @ fx, Bm = sgy @ fy   (fp32 WMMA 16x16x4,
//                     double-buffered GLOBAL_LOAD_ASYNC_TO_LDS_B128 staging)
//   2) gram         : AAH = A A^H, BAH = Bm A^H     (fp32 WMMA 16x16x4)
//   3) resolvent    : D(b,k,j) mask from eigenvalues
//   4) solve        : 1024 x (128x128 complex GE solves), Q = conj(x)
// ---------------------------------------------------------------------------

typedef float v2f __attribute__((ext_vector_type(2)));
typedef float v8f __attribute__((ext_vector_type(8)));

#define LAMBDA 0.001f
#define Bsz 8
#define Nn  8192
#define Kk  128
#define Cc  256

// LDS layout for kernel 1 (per buffer, in floats). Row stride 76:
//  76*4 = 304 bytes (16B aligned for b128 async writes), 76 mod 64 = 12 and
//  12*r mod 64 distinct for r=0..15 -> conflict-free per-lane row reads.
#define SG_STRIDE 76
#define SG_FLOATS (4 * 16 * SG_STRIDE)            // 4864
#define F_FLOATS  (2 * 64 * SG_STRIDE)            // 9728
#define BUF_FLOATS (SG_FLOATS + F_FLOATS)         // 14592 (58368 B per buffer)

// Async copy: 16 bytes global -> LDS via the CDNA5 async path (ASYNCcnt).
__device__ __forceinline__ void async_b128(unsigned lds_addr, const float* g)
{
    asm volatile("global_load_async_to_lds_b128 %0, %1, off"
                 :: "v"(lds_addr), "v"(g) : "memory");
}

// ---------------------------------------------------------------------------
// Kernel 1: spectral projection GEMMs.
// grid = (Cc/64, Kk/16, Bsz), block = 128 (4 waves).
// Each wave computes one 16x16 tile of A_re, A_im, B_re, B_im.
// N is consumed in 64-wide chunks, double-buffered through LDS with async
// loads; s_wait_asynccnt 24 overlaps chunk i+1's DMA with chunk i's WMMAs.
// ---------------------------------------------------------------------------
__global__ __launch_bounds__(128)
void spec_project_kernel(const float* __restrict__ sgxre, const float* __restrict__ sgxim,
                         const float* __restrict__ sgyre, const float* __restrict__ sgyim,
                         const float* __restrict__ fx,    const float* __restrict__ fy,
                         float* __restrict__ Are, float* __restrict__ Aim,
                         float* __restrict__ Bre, float* __restrict__ Bim)
{
    const int b   = blockIdx.z;
    const int m0  = blockIdx.y * 16;
    const int c0g = blockIdx.x * 64;
    const int tid = threadIdx.x;
    const int wave = tid >> 5;
    const int lane = tid & 31;
    const int half = lane >> 4;   // 0: lanes 0-15, 1: lanes 16-31
    const int lrow = lane & 15;
    const int cw   = wave * 16;   // wave's 16-col sub-tile within the 64-col group

    extern __shared__ float smem[];   // 2 * BUF_FLOATS

    v8f accAre = {}; v8f accAim = {}; v8f accBre = {}; v8f accBim = {};

    const float* sgBase[4] = { sgxre + ((size_t)b * Kk + m0) * Nn,
                               sgxim + ((size_t)b * Kk + m0) * Nn,
                               sgyre + ((size_t)b * Kk + m0) * Nn,
                               sgyim + ((size_t)b * Kk + m0) * Nn };
    const float* fBase[2]  = { fx + (size_t)b * Nn * Cc + c0g,
                               fy + (size_t)b * Nn * Cc + c0g };

    // Issue one chunk's staging (24 b128 async ops per thread):
    //   sg tiles : 4 matrices x 16 rows x 64 cols  = 1024 ops / 128 thr = 8
    //   feat     : 2 matrices x 64 rows x 64 cols  = 2048 ops / 128 thr = 16
    auto issue_chunk = [&](int nn0, int bufSel) {
        const int bufBase = bufSel * BUF_FLOATS;
        #pragma unroll
        for (int i = 0; i < 8; ++i) {
            int q   = tid + i * 128;
            int m   = q >> 8;            // 256 ops per sg matrix
            int rem = q & 255;
            int r   = rem >> 4;
            int cq  = (rem & 15) << 2;
            const float* g = sgBase[m] + (size_t)r * Nn + nn0 + cq;
            unsigned lds = (unsigned)(size_t)&smem[bufBase + (m * 16 + r) * SG_STRIDE + cq];
            async_b128(lds, g);
        }
        #pragma unroll
        for (int i = 0; i < 16; ++i) {
            int q   = tid + i * 128;
            int m   = q >> 10;           // 1024 ops per feat matrix
            int rem = q & 1023;
            int r   = rem >> 4;
            int cq  = (rem & 15) << 2;
            const float* g = fBase[m] + (size_t)(nn0 + r) * Cc + cq;
            unsigned lds = (unsigned)(size_t)&smem[bufBase + SG_FLOATS + (m * 64 + r) * SG_STRIDE + cq];
            async_b128(lds, g);
        }
    };

    issue_chunk(0, 0);

    int buf = 0;
    for (int ch = 0; ch < Nn / 64; ++ch) {
        const bool more = (ch + 1) < (Nn / 64);
        if (more) issue_chunk((ch + 1) * 64, buf ^ 1);

        // Async loads complete in order: waiting <=24 guarantees the current
        // chunk's 24 ops are done while next chunk's 24 remain in flight.
        if (more) asm volatile("s_wait_asynccnt 0x18" ::: "memory");
        else      asm volatile("s_wait_asynccnt 0x0"  ::: "memory");
        __syncthreads();

        const float* S = &smem[buf * BUF_FLOATS];
        const float* F = S + SG_FLOATS;

        #pragma unroll 4
        for (int kk = 0; kk < 64; kk += 4) {
            const int ka = kk + 2 * half;
            // A-operand (16x4 f32): lane holds row M=lrow, v0/v1 = K {0,1} or {2,3}
            v2f a_xre = { S[(0 * 16 + lrow) * SG_STRIDE + ka], S[(0 * 16 + lrow) * SG_STRIDE + ka + 1] };
            v2f a_xim = { S[(1 * 16 + lrow) * SG_STRIDE + ka], S[(1 * 16 + lrow) * SG_STRIDE + ka + 1] };
            v2f a_yre = { S[(2 * 16 + lrow) * SG_STRIDE + ka], S[(2 * 16 + lrow) * SG_STRIDE + ka + 1] };
            v2f a_yim = { S[(3 * 16 + lrow) * SG_STRIDE + ka], S[(3 * 16 + lrow) * SG_STRIDE + ka + 1] };
            // B-operand (4x16 f32): lane holds col N=lrow, rows K {0,1}/{2,3}
            v2f b_x = { F[(0 * 64 + ka) * SG_STRIDE + cw + lrow], F[(0 * 64 + ka + 1) * SG_STRIDE + cw + lrow] };
            v2f b_y = { F[(1 * 64 + ka) * SG_STRIDE + cw + lrow], F[(1 * 64 + ka + 1) * SG_STRIDE + cw + lrow] };

            accAre = __builtin_amdgcn_wmma_f32_16x16x4_f32(false, a_xre, false, b_x, (short)0, accAre, false, false);
            accAim = __builtin_amdgcn_wmma_f32_16x16x4_f32(false, a_xim, false, b_x, (short)0, accAim, false, false);
            accBre = __builtin_amdgcn_wmma_f32_16x16x4_f32(false, a_yre, false, b_y, (short)0, accBre, false, false);
            accBim = __builtin_amdgcn_wmma_f32_16x16x4_f32(false, a_yim, false, b_y, (short)0, accBim, false, false);
        }
        __syncthreads();   // done reading buf before it is refilled
        buf ^= 1;
    }

    // C/D layout: VGPR i -> row M = i + 8*half, col N = lrow
    #pragma unroll
    for (int i = 0; i < 8; ++i) {
        int row = m0 + i + 8 * half;
        int col = c0g + cw + lrow;
        size_t off = ((size_t)b * Kk + row) * Cc + col;
        Are[off] = accAre[i];
        Aim[off] = accAim[i];
        Bre[off] = accBre[i];
        Bim[off] = accBim[i];
    }
}

// ---------------------------------------------------------------------------
// Kernel 2: Gram matrices AAH = A A^H, BAH = Bm A^H (complex, reduce over C).
// grid = (Kk/16, Kk/16, Bsz), block = 32 (one wave per 16x16 tile).
// f32 WMMA lacks A/B negate -> keep 8 separate accumulators, combine at end.
// ---------------------------------------------------------------------------
__global__ __launch_bounds__(32)
void gram_kernel(const float* __restrict__ Are, const float* __restrict__ Aim,
                 const float* __restrict__ Bre, const float* __restrict__ Bim,
                 float* __restrict__ AAHre, float* __restrict__ AAHim,
                 float* __restrict__ BAHre, float* __restrict__ BAHim)
{
    const int b  = blockIdx.z;
    const int i0 = blockIdx.y * 16;
    const int j0 = blockIdx.x * 16;
    const int lane = threadIdx.x & 31;
    const int half = lane >> 4;
    const int lrow = lane & 15;

    const float* ArI = Are + ((size_t)b * Kk + i0 + lrow) * Cc;
    const float* AiI = Aim + ((size_t)b * Kk + i0 + lrow) * Cc;
    const float* BrI = Bre + ((size_t)b * Kk + i0 + lrow) * Cc;
    const float* BiI = Bim + ((size_t)b * Kk + i0 + lrow) * Cc;
    const float* ArJ = Are + ((size_t)b * Kk + j0 + lrow) * Cc;
    const float* AiJ = Aim + ((size_t)b * Kk + j0 + lrow) * Cc;

    v8f aRR = {}, aII = {}, aIR = {}, aRI = {};
    v8f bRR = {}, bII = {}, bIR = {}, bRI = {};

    for (int cc = 0; cc < Cc; cc += 4) {
        const int k2 = cc + 2 * half;
        v2f xre = { ArI[k2], ArI[k2 + 1] };
        v2f xim = { AiI[k2], AiI[k2 + 1] };
        v2f yre = { BrI[k2], BrI[k2 + 1] };
        v2f yim = { BiI[k2], BiI[k2 + 1] };
        v2f jre = { ArJ[k2], ArJ[k2 + 1] };   // rows j along C == B-operand of X*Y^T
        v2f jim = { AiJ[k2], AiJ[k2 + 1] };

        aRR = __builtin_amdgcn_wmma_f32_16x16x4_f32(false, xre, false, jre, (short)0, aRR, false, false);
        aII = __builtin_amdgcn_wmma_f32_16x16x4_f32(false, xim, false, jim, (short)0, aII, false, false);
        aIR = __builtin_amdgcn_wmma_f32_16x16x4_f32(false, xim, false, jre, (short)0, aIR, false, false);
        aRI = __builtin_amdgcn_wmma_f32_16x16x4_f32(false, xre, false, jim, (short)0, aRI, false, false);
        bRR = __builtin_amdgcn_wmma_f32_16x16x4_f32(false, yre, false, jre, (short)0, bRR, false, false);
        bII = __builtin_amdgcn_wmma_f32_16x16x4_f32(false, yim, false, jim, (short)0, bII, false, false);
        bIR = __builtin_amdgcn_wmma_f32_16x16x4_f32(false, yim, false, jre, (short)0, bIR, false, false);
        bRI = __builtin_amdgcn_wmma_f32_16x16x4_f32(false, yre, false, jim, (short)0, bRI, false, false);
    }

    #pragma unroll
    for (int i = 0; i < 8; ++i) {
        int row = i0 + i + 8 * half;
        int col = j0 + lrow;
        size_t off = ((size_t)b * Kk + row) * Kk + col;
        AAHre[off] = aRR[i] + aII[i];
        AAHim[off] = aIR[i] - aRI[i];
        BAHre[off] = bRR[i] + bII[i];
        BAHim[off] = bIR[i] - bRI[i];
    }
}

// ---------------------------------------------------------------------------
// Kernel 3: resolvent mask D[b,k,j]. grid = Bsz, block = Kk.
// gamma = 0.5 -> sqrt.
// ---------------------------------------------------------------------------
__global__ __launch_bounds__(128)
void resolvent_kernel(const float* __restrict__ cevx, const float* __restrict__ cevy,
                      float* __restrict__ Dm)
{
    const int b = blockIdx.x;
    const int t = threadIdx.x;
    __shared__ float ex[Kk], ey[Kk], red[Kk];

    float vx = cevx[b * Kk + t];
    float vy = cevy[b * Kk + t];
    ex[t] = vx;
    ey[t] = vy;
    red[t] = fmaxf(vx, vy);
    __syncthreads();
    for (int s = Kk / 2; s > 0; s >>= 1) {
        if (t < s) red[t] = fmaxf(red[t], red[t + s]);
        __syncthreads();
    }
    const float sMax = red[0];
    const float g1 = sqrtf(ex[t] / sMax);
    const float d1 = g1 * g1 + 1.0f;
    const float h1 = g1 / d1;
    const float e1 = 1.0f / d1;
    for (int k = 0; k < Kk; ++k) {
        float g2 = sqrtf(ey[k] / sMax);
        float d2 = g2 * g2 + 1.0f;
        float mre = g2 / d2 - h1;
        float mim = 1.0f / d2 - e1;
        Dm[((size_t)b * Kk + k) * Kk + t] = mre * mre + mim * mim;
    }
}

// ---------------------------------------------------------------------------
// Kernel 4: per-(b,k) 128x128 complex solve (HPD -> GE without pivoting).
// grid = (Kk, Bsz), block = Kk.  Dynamic LDS: 128x128 float2 matrix + rhs
// (~129 KB; legal on CDNA5's 320 KB/WGP LDS).
// ---------------------------------------------------------------------------
__global__ __launch_bounds__(128)
void solve_kernel(const float* __restrict__ AAHre, const float* __restrict__ AAHim,
                  const float* __restrict__ BAHre, const float* __restrict__ BAHim,
                  const float* __restrict__ Dm, double* __restrict__ Q)
{
    const int k = blockIdx.x;
    const int b = blockIdx.y;
    const int r = threadIdx.x;

    extern __shared__ float2 sm[];
    float2* M   = sm;             // [Kk*Kk]
    float2* rhs = sm + Kk * Kk;   // [Kk]

    // M = AAH[b] + LAMBDA * diag(D[b,k,:]) ; rhs = conj(BAH[b,k,:])
    const float* ar = AAHre + ((size_t)b * Kk + r) * Kk;
    const float* ai = AAHim + ((size_t)b * Kk + r) * Kk;
    for (int c = 0; c < Kk; ++c) {
        float re = ar[c];
        if (c == r) re += LAMBDA * Dm[((size_t)b * Kk + k) * Kk + r];
        M[r * Kk + c] = make_float2(re, ai[c]);
    }
    rhs[r] = make_float2( BAHre[((size_t)b * Kk + k) * Kk + r],
                         -BAHim[((size_t)b * Kk + k) * Kk + r]);
    __syncthreads();

    // Forward elimination (thread r owns row r)
    for (int p = 0; p < Kk - 1; ++p) {
        float2 piv = M[p * Kk + p];
        float inv = 1.0f / (piv.x * piv.x + piv.y * piv.y);
        if (r > p) {
            float2 mrp = M[r * Kk + p];
            float fr = (mrp.x * piv.x + mrp.y * piv.y) * inv;
            float fi = (mrp.y * piv.x - mrp.x * piv.y) * inv;
            for (int c = p + 1; c < Kk; ++c) {
                float2 mp = M[p * Kk + c];
                float2 mr = M[r * Kk + c];
                mr.x -= fr * mp.x - fi * mp.y;
                mr.y -= fr * mp.y + fi * mp.x;
                M[r * Kk + c] = mr;
            }
            float2 rp = rhs[p];
            float2 rr = rhs[r];
            rr.x -= fr * rp.x - fi * rp.y;
            rr.y -= fr * rp.y + fi * rp.x;
            rhs[r] = rr;
        }
        __syncthreads();
    }

    // Back substitution
    for (int p = Kk - 1; p >= 0; --p) {
        if (r == p) {
            float2 piv = M[p * Kk + p];
            float inv = 1.0f / (piv.x * piv.x + piv.y * piv.y);
            float2 rp = rhs[p];
            rhs[p] = make_float2((rp.x * piv.x + rp.y * piv.y) * inv,
                                 (rp.y * piv.x - rp.x * piv.y) * inv);
        }
        __syncthreads();
        if (r < p) {
            float2 xp = rhs[p];
            float2 mrp = M[r * Kk + p];
            float2 rr = rhs[r];
            rr.x -= mrp.x * xp.x - mrp.y * xp.y;
            rr.y -= mrp.x * xp.y + mrp.y * xp.x;
            rhs[r] = rr;
        }
        __syncthreads();
    }

    // Q[b,k,r] = conj(x[r]) as complex128 (interleaved double re, im)
    size_t off = ((size_t)b * Kk + k) * Kk + r;
    Q[2 * off]     = (double)rhs[r].x;
    Q[2 * off + 1] = (double)(-rhs[r].y);
}

// ---------------------------------------------------------------------------
extern "C" void kernel_launch(void* const* d_in, const int* in_sizes, int n_in,
                              void* d_out, int out_size, void* d_ws, size_t ws_size,
                              hipStream_t stream)
{
    const float* fx    = (const float*)d_in[0];
    const float* fy    = (const float*)d_in[1];
    const float* sgxre = (const float*)d_in[2];
    const float* sgxim = (const float*)d_in[3];
    const float* sgyre = (const float*)d_in[4];
    const float* sgyim = (const float*)d_in[5];
    const float* cevx  = (const float*)d_in[6];
    const float* cevy  = (const float*)d_in[7];
    double* Q = (double*)d_out;

    // Workspace layout (floats): 4x B*K*C + 4x B*K*K + B*K*K  ~= 6.5 MB
    float* ws = (float*)d_ws;
    const size_t nAC = (size_t)Bsz * Kk * Cc;   // 262144
    const size_t nKK = (size_t)Bsz * Kk * Kk;   // 131072
    float* Are   = ws;
    float* Aim   = Are + nAC;
    float* Bre   = Aim + nAC;
    float* Bim   = Bre + nAC;
    float* AAHre = Bim + nAC;
    float* AAHim = AAHre + nKK;
    float* BAHre = AAHim + nKK;
    float* BAHim = BAHre + nKK;
    float* Dm    = BAHim + nKK;

    dim3 g1(Cc / 64, Kk / 16, Bsz);   // (4, 8, 8)
    size_t smem1 = (size_t)2 * BUF_FLOATS * sizeof(float);   // 116736 B
    spec_project_kernel<<<g1, 128, smem1, stream>>>(sgxre, sgxim, sgyre, sgyim, fx, fy,
                                                    Are, Aim, Bre, Bim);

    dim3 g2(Kk / 16, Kk / 16, Bsz);   // (8, 8, 8)
    gram_kernel<<<g2, 32, 0, stream>>>(Are, Aim, Bre, Bim, AAHre, AAHim, BAHre, BAHim);

    resolvent_kernel<<<Bsz, Kk, 0, stream>>>(cevx, cevy, Dm);

    dim3 g3(Kk, Bsz);                 // 1024 solves
    size_t smem3 = (size_t)(Kk * Kk + Kk) * sizeof(float2);
    solve_kernel<<<g3, Kk, smem3, stream>>>(AAHre, AAHim, BAHre, BAHim, Dm, Q);
}